// GAT_actor_55327768708411
// MI455X (gfx1250) — compile-verified
//
#include <hip/hip_runtime.h>
#include <math.h>

// GAT actor, 2-layer (H1=4,NHID=64) -> (H2=1,NCLASS=16), fp32 end to end.
// Matrix ops use CDNA5 V_WMMA_F32_16X16X4_F32 (wave32).

#define NFEAT   128
#define NHID    64
#define H1N     4
#define NCLASS  16
#define F1      (H1N * NHID)   // 256
#define NEG_SLOPE 0.2f

typedef float v2f __attribute__((ext_vector_type(2)));
typedef float v8f __attribute__((ext_vector_type(8)));

// ---- order-preserving float <-> u32 key (for atomicMax on floats) ----
__device__ __forceinline__ unsigned fkey(float f) {
  unsigned u = __float_as_uint(f);
  return (u & 0x80000000u) ? ~u : (u | 0x80000000u);
}
__device__ __forceinline__ float fdec(unsigned u) {
  return __uint_as_float((u & 0x80000000u) ? (u ^ 0x80000000u) : ~u);
}
#define NEGINF_KEY 0x007FFFFFu   // fkey(-inf)

// ---------------- fills ----------------
__global__ void fill_u32(unsigned* __restrict__ p, unsigned v, int n) {
  int i = blockIdx.x * blockDim.x + threadIdx.x;
  if (i < n) p[i] = v;
}
__global__ void fill_f32(float* __restrict__ p, float v, long long n) {
  long long i = (long long)blockIdx.x * blockDim.x + threadIdx.x;
  if (i < n) p[i] = v;
}

// ---------------- f32 WMMA GEMM: C[MxNc] = A[MxK] * B[KxNc] ----------------
// One wave (32 threads) per 16x16 output tile. Requires M%16==0, Nc%16==0, K%4==0.
__global__ void wmma_gemm_f32(const float* __restrict__ A, int lda,
                              const float* __restrict__ B, int ldb,
                              float* __restrict__ C, int ldc, int K) {
  const int lane = threadIdx.x & 31;
  const int row0 = blockIdx.x * 16;
  const int col0 = blockIdx.y * 16;
  const int mn = lane & 15;            // row of A / col of B this lane holds
  const int kb = (lane >> 4) * 2;      // lane-half K offset (f32 16x16x4 layout)

  v8f acc = {};
  for (int k0 = 0; k0 < K; k0 += 4) {
    v2f a, b;
    a.x = A[(size_t)(row0 + mn) * lda + (k0 + kb + 0)];
    a.y = A[(size_t)(row0 + mn) * lda + (k0 + kb + 1)];
    b.x = B[(size_t)(k0 + kb + 0) * ldb + (col0 + mn)];
    b.y = B[(size_t)(k0 + kb + 1) * ldb + (col0 + mn)];
    acc = __builtin_amdgcn_wmma_f32_16x16x4_f32(
        /*neg_a=*/false, a, /*neg_b=*/false, b,
        /*c_mod=*/(short)0, acc, /*reuse_a=*/false, /*reuse_b=*/false);
  }
  // C/D layout: VGPR j -> row (j + 8*(lane>>4)), col (lane&15)
  const int rbase = row0 + ((lane >> 4) * 8);
#pragma unroll
  for (int j = 0; j < 8; ++j)
    C[(size_t)(rbase + j) * ldc + (col0 + mn)] = acc[j];
}

// ---------------- per-node attention logits ----------------
// asrc[n*H+h] = dot(xw[n, h*C : (h+1)*C], att_src[h]);  same for adst.
__global__ void att_logits(const float* __restrict__ xw,
                           const float* __restrict__ att_src,
                           const float* __restrict__ att_dst,
                           float* __restrict__ asrc, float* __restrict__ adst,
                           int n, int H, int C) {
  int idx = blockIdx.x * blockDim.x + threadIdx.x;
  if (idx >= n * H) return;
  int node = idx / H, h = idx % H;
  const float* xp = xw + (size_t)node * H * C + (size_t)h * C;
  const float* ws = att_src + (size_t)h * C;
  const float* wd = att_dst + (size_t)h * C;
  float s = 0.f, d = 0.f;
  for (int c = 0; c < C; ++c) {
    float v = xp[c];
    s = fmaf(v, ws[c], s);
    d = fmaf(v, wd[c], d);
  }
  asrc[idx] = s;
  adst[idx] = d;
}

// ---------------- edge pass 1: leaky-relu logit + segment max ----------------
// Edges e in [0,E) come from edge_index; e in [E, E+n) are self loops (i,i).
__global__ void edge_max(const int* __restrict__ ei, int E, int n,
                         const float* __restrict__ asrc,
                         const float* __restrict__ adst,
                         unsigned* __restrict__ mkey, float* __restrict__ eb,
                         int H) {
  int idx = blockIdx.x * blockDim.x + threadIdx.x;
  int tot = (E + n) * H;
  if (idx >= tot) return;
  int e = idx / H, h = idx % H;
  int s = (e < E) ? ei[e] : (e - E);
  int d = (e < E) ? ei[E + e] : (e - E);
  float v = asrc[s * H + h] + adst[d * H + h];
  v = (v > 0.f) ? v : NEG_SLOPE * v;
  eb[idx] = v;
  atomicMax(&mkey[d * H + h], fkey(v));
}

// ---------------- edge pass 2: exp(e - max) + segment sum ----------------
__global__ void edge_expsum(const int* __restrict__ ei, int E, int n,
                            const unsigned* __restrict__ mkey,
                            float* __restrict__ eb, float* __restrict__ ssum,
                            int H) {
  int idx = blockIdx.x * blockDim.x + threadIdx.x;
  int tot = (E + n) * H;
  if (idx >= tot) return;
  int e = idx / H, h = idx % H;
  int d = (e < E) ? ei[E + e] : (e - E);
  float m = fdec(mkey[d * H + h]);
  float ex = __expf(eb[idx] - m);
  eb[idx] = ex;
  atomicAdd(&ssum[d * H + h], ex);
}

// ---------------- edge pass 3: weighted scatter out[dst] += alpha * xw[src] --
// One thread per (edge, feature). F = H*C features. xw1/out both L2-resident.
__global__ void edge_scatter(const int* __restrict__ ei, int E, int n,
                             const float* __restrict__ eb,
                             const float* __restrict__ ssum,
                             const float* __restrict__ xw,
                             float* __restrict__ out, int H, int C) {
  long long idx = (long long)blockIdx.x * blockDim.x + threadIdx.x;
  const int F = H * C;
  long long tot = (long long)(E + n) * F;
  if (idx >= tot) return;
  int e = (int)(idx / F);
  int f = (int)(idx % F);
  int h = f / C;
  int s = (e < E) ? ei[e] : (e - E);
  int d = (e < E) ? ei[E + e] : (e - E);
  float alpha = eb[(size_t)e * H + h] / ssum[(size_t)d * H + h];
  atomicAdd(&out[(size_t)d * F + f], alpha * xw[(size_t)s * F + f]);
}

// ---------------- bias + ELU (in place) ----------------
__global__ void bias_elu(float* __restrict__ x, const float* __restrict__ bias,
                         long long total, int F) {
  long long idx = (long long)blockIdx.x * blockDim.x + threadIdx.x;
  if (idx >= total) return;
  float v = x[idx] + bias[(int)(idx % F)];
  x[idx] = (v > 0.f) ? v : expm1f(v);
}

// ---------------- bias + row softmax over 16 classes ----------------
__global__ void bias_softmax16(const float* __restrict__ acc,
                               const float* __restrict__ bias,
                               float* __restrict__ out, int n) {
  int node = blockIdx.x * blockDim.x + threadIdx.x;
  if (node >= n) return;
  float v[NCLASS];
  float m = -INFINITY;
#pragma unroll
  for (int i = 0; i < NCLASS; ++i) {
    v[i] = acc[(size_t)node * NCLASS + i] + bias[i];
    m = fmaxf(m, v[i]);
  }
  float s = 0.f;
#pragma unroll
  for (int i = 0; i < NCLASS; ++i) { v[i] = __expf(v[i] - m); s += v[i]; }
  float inv = 1.f / s;
#pragma unroll
  for (int i = 0; i < NCLASS; ++i)
    out[(size_t)node * NCLASS + i] = v[i] * inv;
}

static inline unsigned cdiv(long long a, int b) { return (unsigned)((a + b - 1) / b); }

extern "C" void kernel_launch(void* const* d_in, const int* in_sizes, int n_in,
                              void* d_out, int out_size, void* d_ws, size_t ws_size,
                              hipStream_t stream) {
  const float* obs      = (const float*)d_in[0];
  const int*   ei       = (const int*)  d_in[1];   // [2,E] row-major: src row then dst row
  const float* W1       = (const float*)d_in[2];   // [128,256]
  const float* att_src1 = (const float*)d_in[3];   // [4,64]
  const float* att_dst1 = (const float*)d_in[4];
  const float* bias1    = (const float*)d_in[5];   // [256]
  const float* W2       = (const float*)d_in[6];   // [256,16]
  const float* att_src2 = (const float*)d_in[7];   // [1,16]
  const float* att_dst2 = (const float*)d_in[8];
  const float* bias2    = (const float*)d_in[9];   // [16]

  const int N  = in_sizes[0] / NFEAT;   // 50000 (multiple of 16)
  const int E  = in_sizes[1] / 2;       // 800000
  const int ET = E + N;                 // edges incl. self loops

  // ---- workspace carve-out (fp32 elements) ----
  float* p = (float*)d_ws;
  float*    xw1   = p; p += (size_t)N * F1;       // 12.8M
  float*    out1  = p; p += (size_t)N * F1;       // 12.8M (becomes x2 after ELU)
  float*    asrc1 = p; p += (size_t)N * H1N;
  float*    adst1 = p; p += (size_t)N * H1N;
  unsigned* mkey1 = (unsigned*)p; p += (size_t)N * H1N;
  float*    s1    = p; p += (size_t)N * H1N;
  float*    eb1   = p; p += (size_t)ET * H1N;     // 3.4M
  float*    xw2   = p; p += (size_t)N * NCLASS;
  float*    asrc2 = p; p += (size_t)N;
  float*    adst2 = p; p += (size_t)N;
  unsigned* mkey2 = (unsigned*)p; p += (size_t)N;
  float*    s2    = p; p += (size_t)N;
  float*    eb2   = p; p += (size_t)ET;
  float*    out2  = p; p += (size_t)N * NCLASS;

  const int B = 256;

  // ---- init accumulators ----
  fill_u32<<<cdiv((long long)N * H1N, B), B, 0, stream>>>(mkey1, NEGINF_KEY, N * H1N);
  fill_u32<<<cdiv(N, B), B, 0, stream>>>(mkey2, NEGINF_KEY, N);
  fill_f32<<<cdiv((long long)N * H1N, B), B, 0, stream>>>(s1, 0.f, (long long)N * H1N);
  fill_f32<<<cdiv(N, B), B, 0, stream>>>(s2, 0.f, (long long)N);
  fill_f32<<<cdiv((long long)N * F1, B), B, 0, stream>>>(out1, 0.f, (long long)N * F1);
  fill_f32<<<cdiv((long long)N * NCLASS, B), B, 0, stream>>>(out2, 0.f, (long long)N * NCLASS);

  // ---- layer 1 ----
  wmma_gemm_f32<<<dim3(N / 16, F1 / 16), 32, 0, stream>>>(obs, NFEAT, W1, F1, xw1, F1, NFEAT);
  att_logits<<<cdiv((long long)N * H1N, B), B, 0, stream>>>(xw1, att_src1, att_dst1,
                                                            asrc1, adst1, N, H1N, NHID);
  edge_max<<<cdiv((long long)ET * H1N, B), B, 0, stream>>>(ei, E, N, asrc1, adst1, mkey1, eb1, H1N);
  edge_expsum<<<cdiv((long long)ET * H1N, B), B, 0, stream>>>(ei, E, N, mkey1, eb1, s1, H1N);
  edge_scatter<<<cdiv((long long)ET * F1, B), B, 0, stream>>>(ei, E, N, eb1, s1, xw1, out1, H1N, NHID);
  bias_elu<<<cdiv((long long)N * F1, B), B, 0, stream>>>(out1, bias1, (long long)N * F1, F1);

  // ---- layer 2 ----
  wmma_gemm_f32<<<dim3(N / 16, NCLASS / 16), 32, 0, stream>>>(out1, F1, W2, NCLASS, xw2, NCLASS, F1);
  att_logits<<<cdiv(N, B), B, 0, stream>>>(xw2, att_src2, att_dst2, asrc2, adst2, N, 1, NCLASS);
  edge_max<<<cdiv((long long)ET, B), B, 0, stream>>>(ei, E, N, asrc2, adst2, mkey2, eb2, 1);
  edge_expsum<<<cdiv((long long)ET, B), B, 0, stream>>>(ei, E, N, mkey2, eb2, s2, 1);
  edge_scatter<<<cdiv((long long)ET * NCLASS, B), B, 0, stream>>>(ei, E, N, eb2, s2, xw2, out2, 1, NCLASS);
  bias_softmax16<<<cdiv(N, B), B, 0, stream>>>(out2, bias2, (float*)d_out, N);
}